// BiAlignAggLayer_32246614458531
// MI455X (gfx1250) — compile-verified
//
#include <hip/hip_runtime.h>

// Problem constants (match reference)
constexpr int BB = 32;    // batch
constexpr int L  = 512;   // Li == Lj
constexpr int D  = 512;   // feature dim
constexpr int H  = 512;   // 2*NN_DIM

typedef __attribute__((ext_vector_type(2))) float v2f;
typedef __attribute__((ext_vector_type(8))) float v8f;

__device__ __forceinline__ v8f wmma_f32_16x16x4(v2f a, v2f b, v8f c) {
  // D = A(16x4 f32) * B(4x16 f32) + C(16x16 f32), full fp32 matrix path
  return __builtin_amdgcn_wmma_f32_16x16x4_f32(false, a, false, b, (short)0, c,
                                               false, false);
}

__device__ __forceinline__ v8f v8f_zero() {
  v8f v;
#pragma unroll
  for (int i = 0; i < 8; ++i) v[i] = 0.0f;
  return v;
}

// -------------------------------------------------------------------------
// Kernel 1: S[b,i,j] = sum_k I[b,i,k] * J[b,j,k]
// 128x128 tile per 256-thread block; 8 waves (2 M x 4 N), wave tile 64x32.
// -------------------------------------------------------------------------
__global__ __launch_bounds__(256) void k_scores(const float* __restrict__ Iin,
                                                const float* __restrict__ Jin,
                                                float* __restrict__ S) {
  const int b  = blockIdx.z;
  const int i0 = blockIdx.y * 128;
  const int j0 = blockIdx.x * 128;
  const float* Ib = Iin + (size_t)b * L * D;
  const float* Jb = Jin + (size_t)b * L * D;
  float* Sb = S + (size_t)b * L * L;

  __shared__ float As[128][36];  // pad 36: conflict-free strided reads, 16B-aligned rows
  __shared__ float Bs[128][36];

  const int tid  = threadIdx.x;
  const int lane = tid & 31;
  const int lm   = lane & 15;          // M/N index inside 16-tile
  const int lk   = (lane >> 4) * 2;    // K sub-pair selector (A/B f32 layout)
  const int wave = tid >> 5;
  const int wm0  = (wave >> 2) * 64;   // 2 waves along M
  const int wn0  = (wave & 3) * 32;    // 4 waves along N

  v8f acc[4][2];
#pragma unroll
  for (int m = 0; m < 4; ++m)
#pragma unroll
    for (int n = 0; n < 2; ++n) acc[m][n] = v8f_zero();

  const int lr = tid >> 3;        // 0..31 staging row
  const int lc = (tid & 7) * 4;   // 0..28 staging col (float4)

  for (int k0 = 0; k0 < D; k0 += 32) {
    __syncthreads();
#pragma unroll
    for (int it = 0; it < 4; ++it) {
      const int r = lr + it * 32;
      const float4 av = *(const float4*)(Ib + (size_t)(i0 + r) * D + k0 + lc);
      const float4 bv = *(const float4*)(Jb + (size_t)(j0 + r) * D + k0 + lc);
      *(float4*)&As[r][lc] = av;
      *(float4*)&Bs[r][lc] = bv;
    }
    __syncthreads();
#pragma unroll
    for (int kk = 0; kk < 32; kk += 4) {
      v2f a[4], bv2[2];
#pragma unroll
      for (int mt = 0; mt < 4; ++mt) {
        a[mt].x = As[wm0 + mt * 16 + lm][kk + lk];
        a[mt].y = As[wm0 + mt * 16 + lm][kk + lk + 1];
      }
#pragma unroll
      for (int nt = 0; nt < 2; ++nt) {
        bv2[nt].x = Bs[wn0 + nt * 16 + lm][kk + lk];
        bv2[nt].y = Bs[wn0 + nt * 16 + lm][kk + lk + 1];
      }
#pragma unroll
      for (int mt = 0; mt < 4; ++mt)
#pragma unroll
        for (int nt = 0; nt < 2; ++nt)
          acc[mt][nt] = wmma_f32_16x16x4(a[mt], bv2[nt], acc[mt][nt]);
    }
  }

  const int rh = (lane >> 4) * 8;  // C/D layout: lanes 16-31 hold M = r+8
#pragma unroll
  for (int mt = 0; mt < 4; ++mt)
#pragma unroll
    for (int nt = 0; nt < 2; ++nt)
#pragma unroll
      for (int r = 0; r < 8; ++r)
        Sb[(size_t)(i0 + wm0 + mt * 16 + r + rh) * L + j0 + wn0 + nt * 16 + lm] =
            acc[mt][nt][r];
}

// -------------------------------------------------------------------------
// Kernel 2: row softmax stats: max_j and 1/sum_j exp(S - max) per (b,i)
// One wave per row, 8 rows per block.
// -------------------------------------------------------------------------
__global__ __launch_bounds__(256) void k_rowstats(const float* __restrict__ S,
                                                  float* __restrict__ rmax,
                                                  float* __restrict__ rinv) {
  const int row  = blockIdx.x * 8 + (threadIdx.x >> 5);  // global row in [0, B*L)
  const int lane = threadIdx.x & 31;
  const float* Sr = S + (size_t)row * L;
  float m = -1e30f;
  for (int j = lane; j < L; j += 32) m = fmaxf(m, Sr[j]);
#pragma unroll
  for (int off = 16; off > 0; off >>= 1) m = fmaxf(m, __shfl_xor(m, off, 32));
  float s = 0.0f;
  for (int j = lane; j < L; j += 32) s += __expf(Sr[j] - m);
#pragma unroll
  for (int off = 16; off > 0; off >>= 1) s += __shfl_xor(s, off, 32);
  if (lane == 0) {
    rmax[row] = m;
    rinv[row] = 1.0f / s;
  }
}

// -------------------------------------------------------------------------
// Kernel 3: column softmax stats per (b,j); 64 columns x 4-way row split.
// -------------------------------------------------------------------------
__global__ __launch_bounds__(256) void k_colstats(const float* __restrict__ S,
                                                  float* __restrict__ cmax,
                                                  float* __restrict__ cinv) {
  const int b  = blockIdx.y;
  const int jt = threadIdx.x & 63;
  const int it = threadIdx.x >> 6;  // 0..3
  const int j  = blockIdx.x * 64 + jt;
  const float* Sb = S + (size_t)b * L * L;
  __shared__ float red[4][64];

  float m = -1e30f;
  for (int i = it * 128; i < it * 128 + 128; ++i)
    m = fmaxf(m, Sb[(size_t)i * L + j]);
  red[it][jt] = m;
  __syncthreads();
  m = fmaxf(fmaxf(red[0][jt], red[1][jt]), fmaxf(red[2][jt], red[3][jt]));
  __syncthreads();
  float s = 0.0f;
  for (int i = it * 128; i < it * 128 + 128; ++i)
    s += __expf(Sb[(size_t)i * L + j] - m);
  red[it][jt] = s;
  __syncthreads();
  if (it == 0) {
    s = red[0][jt] + red[1][jt] + red[2][jt] + red[3][jt];
    cmax[(size_t)b * L + j] = m;
    cinv[(size_t)b * L + j] = 1.0f / s;
  }
}

// -------------------------------------------------------------------------
// Kernel 4 (templated): fully fused
//   TR=false: rows are i, P = rowsoftmax(S), w = P*J, out = |I - w|
//   TR=true : rows are j, P = colsoftmax(S) read transposed, w = P^T*I, out = |J - w|
// Then: agg = out @ W_agg + b, tanh, sum over the 64 rows -> per-block partial.
// 64 rows per block; 8 waves each own a 64-wide D (and later H) slice.
// -------------------------------------------------------------------------
template <bool TR>
__global__ __launch_bounds__(256) void k_fused(
    const float* __restrict__ S,     // [B][L][L]
    const float* __restrict__ Vals,  // aggregated matrix  [B][L][D]
    const float* __restrict__ Res,   // residual matrix    [B][L][D]
    const float* __restrict__ smax,  // [B][L] softmax max (row or col)
    const float* __restrict__ sinv,  // [B][L] 1/sumexp
    const float* __restrict__ Wagg,  // [D][H]
    const float* __restrict__ bagg,  // [H]
    float* __restrict__ Part,        // [16][B][H] partial sums
    int slotBase) {
  const int b    = blockIdx.y;
  const int row0 = blockIdx.x * 64;
  const int tid  = threadIdx.x;
  const int lane = tid & 31;
  const int wave = tid >> 5;
  const int lm   = lane & 15;
  const int lk   = (lane >> 4) * 2;
  const int d0   = wave * 64;  // this wave's D-slice (phase A) / H-slice (phase B)

  const float* Sb = S + (size_t)b * L * L;
  const float* Vb = Vals + (size_t)b * L * D;
  const float* Rb = Res + (size_t)b * L * D;
  const float* mx = smax + (size_t)b * L;
  const float* iv = sinv + (size_t)b * L;

  __shared__ float P[64][36];   // exp(S - max) tile: 64 rows x 32 k
  __shared__ float Xs[64][68];  // |res - w| chunk: 64 rows x 64 d (pad 68: conflict-free)

  v8f acc[4][4];
#pragma unroll
  for (int m = 0; m < 4; ++m)
#pragma unroll
    for (int n = 0; n < 4; ++n) acc[m][n] = v8f_zero();

  // ---- Phase A: w[row, d0..d0+63] = sum_k P[row,k] * Vals[k, d] ----
  for (int k0 = 0; k0 < L; k0 += 32) {
    __syncthreads();
    if (!TR) {
      const int c  = tid & 31;
      const int r0 = tid >> 5;
#pragma unroll
      for (int rr = 0; rr < 8; ++rr) {
        const int r = r0 + rr * 8;
        P[r][c] = __expf(Sb[(size_t)(row0 + r) * L + k0 + c] - mx[row0 + r]);
      }
    } else {
      const int jj  = tid & 63;
      const int ib  = tid >> 6;  // 0..3
      const float cm = mx[row0 + jj];
#pragma unroll
      for (int t = 0; t < 8; ++t) {
        const int ii = ib + t * 4;
        // coalesced read along j, transposed store into P
        P[jj][ii] = __expf(Sb[(size_t)(k0 + ii) * L + row0 + jj] - cm);
      }
    }
    __syncthreads();
#pragma unroll
    for (int kk = 0; kk < 32; kk += 4) {
      v2f a[4], bv[4];
#pragma unroll
      for (int mt = 0; mt < 4; ++mt) {
        a[mt].x = P[mt * 16 + lm][kk + lk];
        a[mt].y = P[mt * 16 + lm][kk + lk + 1];
      }
#pragma unroll
      for (int nt = 0; nt < 4; ++nt) {
        const float* vp =
            Vb + (size_t)(k0 + kk + lk) * D + d0 + nt * 16 + lm;
        bv[nt].x = vp[0];
        bv[nt].y = vp[D];
      }
#pragma unroll
      for (int mt = 0; mt < 4; ++mt)
#pragma unroll
        for (int nt = 0; nt < 4; ++nt)
          acc[mt][nt] = wmma_f32_16x16x4(a[mt], bv[nt], acc[mt][nt]);
    }
  }

  // ---- Phase B: out = |Res - w/sum|, immediately GEMM'd against W_agg ----
  v8f agg[4][4];
#pragma unroll
  for (int m = 0; m < 4; ++m)
#pragma unroll
    for (int n = 0; n < 4; ++n) agg[m][n] = v8f_zero();

  for (int cc = 0; cc < 8; ++cc) {
    __syncthreads();
    if (wave == cc) {  // owner wave publishes its 64x64 |res - w| chunk
#pragma unroll
      for (int mt = 0; mt < 4; ++mt)
#pragma unroll
        for (int r = 0; r < 8; ++r) {
          const int mrow = mt * 16 + r + ((lane >> 4) << 3);
          const float is = iv[row0 + mrow];
#pragma unroll
          for (int nt = 0; nt < 4; ++nt) {
            const int dd = nt * 16 + lm;
            const float w = acc[mt][nt][r] * is;
            Xs[mrow][dd] =
                fabsf(Rb[(size_t)(row0 + mrow) * D + d0 + dd] - w);
          }
        }
    }
    __syncthreads();
#pragma unroll
    for (int kk = 0; kk < 64; kk += 4) {
      v2f a[4], bw[4];
#pragma unroll
      for (int mt = 0; mt < 4; ++mt) {
        a[mt].x = Xs[mt * 16 + lm][kk + lk];
        a[mt].y = Xs[mt * 16 + lm][kk + lk + 1];
      }
#pragma unroll
      for (int nt = 0; nt < 4; ++nt) {
        const float* wp =
            Wagg + (size_t)(cc * 64 + kk + lk) * H + d0 + nt * 16 + lm;
        bw[nt].x = wp[0];
        bw[nt].y = wp[H];
      }
#pragma unroll
      for (int mt = 0; mt < 4; ++mt)
#pragma unroll
        for (int nt = 0; nt < 4; ++nt)
          agg[mt][nt] = wmma_f32_16x16x4(a[mt], bw[nt], agg[mt][nt]);
    }
  }

  // ---- bias + tanh + sum over the 64 rows -> deterministic partial ----
#pragma unroll
  for (int nt = 0; nt < 4; ++nt) {
    const int h    = d0 + nt * 16 + lm;
    const float bz = bagg[h];
    float part = 0.0f;
#pragma unroll
    for (int mt = 0; mt < 4; ++mt)
#pragma unroll
      for (int r = 0; r < 8; ++r) part += tanhf(agg[mt][nt][r] + bz);
    part += __shfl_xor(part, 16, 32);  // combine the two half-wave row groups
    if (lane < 16)
      Part[((size_t)(slotBase + blockIdx.x) * BB + b) * H + h] = part;
  }
}

// -------------------------------------------------------------------------
// Kernel 5: deterministic reduction of the 16 partials, apply mean * 0.5
// -------------------------------------------------------------------------
__global__ __launch_bounds__(256) void k_reduce(const float* __restrict__ Part,
                                                float* __restrict__ Out) {
  const int idx = blockIdx.x * 256 + threadIdx.x;  // over B*H
  float s = 0.0f;
#pragma unroll
  for (int p = 0; p < 16; ++p) s += Part[(size_t)p * BB * H + idx];
  Out[idx] = s * (0.5f / (float)L);
}

// -------------------------------------------------------------------------
extern "C" void kernel_launch(void* const* d_in, const int* in_sizes, int n_in,
                              void* d_out, int out_size, void* d_ws,
                              size_t ws_size, hipStream_t stream) {
  const float* Iin  = (const float*)d_in[0];  // [B, Li, D]
  const float* Jin  = (const float*)d_in[1];  // [B, Lj, D]
  const float* Wagg = (const float*)d_in[2];  // [D, H]
  const float* bagg = (const float*)d_in[3];  // [H]
  float* Out = (float*)d_out;                 // [B, H]

  float* S    = (float*)d_ws;                       // 33.5 MB scores
  float* rmax = S + (size_t)BB * L * L;
  float* rinv = rmax + (size_t)BB * L;
  float* cmax = rinv + (size_t)BB * L;
  float* cinv = cmax + (size_t)BB * L;
  float* Part = cinv + (size_t)BB * L;              // [16][B][H] = 1 MB

  k_scores<<<dim3(L / 128, L / 128, BB), 256, 0, stream>>>(Iin, Jin, S);
  k_rowstats<<<BB * L / 8, 256, 0, stream>>>(S, rmax, rinv);
  k_colstats<<<dim3(L / 64, BB), 256, 0, stream>>>(S, cmax, cinv);
  // output_i path: rows = i, values = J, residual = I, row softmax
  k_fused<false><<<dim3(L / 64, BB), 256, 0, stream>>>(S, Jin, Iin, rmax, rinv,
                                                       Wagg, bagg, Part, 0);
  // output_j path: rows = j, values = I, residual = J, col softmax
  k_fused<true><<<dim3(L / 64, BB), 256, 0, stream>>>(S, Iin, Jin, cmax, cinv,
                                                      Wagg, bagg, Part, 8);
  k_reduce<<<BB * H / 256, 256, 0, stream>>>(Part, Out);
}